// NeRF_83502754169249
// MI455X (gfx1250) — compile-verified
//
#include <hip/hip_runtime.h>
#include <cstdint>

// ---------------------------------------------------------------------------
// NeRF volume rendering, MI455X (gfx1250).
// Memory-bound (~203 MB @ 23.3 TB/s => ~8.7us). No GEMM structure => no WMMA.
// CDNA5 path used: Tensor Data Mover (tensor_load_to_lds) double-buffered
// pipeline for the strided rgb stream + s_wait_tensorcnt, NT cache hints for
// the read-once / write-once streams, wave32-native sizing.
// ---------------------------------------------------------------------------

#define FARD 1e10f
#define EPSV 1e-10f

typedef unsigned int u32x4 __attribute__((ext_vector_type(4)));
typedef int          i32x4 __attribute__((ext_vector_type(4)));
typedef int          i32x8 __attribute__((ext_vector_type(8)));
typedef float        f32x4 __attribute__((ext_vector_type(4)));

#if __has_builtin(__builtin_amdgcn_tensor_load_to_lds) && \
    __has_builtin(__builtin_amdgcn_s_wait_tensorcnt)
#define USE_TDM 1
#else
#define USE_TDM 0
#endif

constexpr int NS   = 64;        // samples per ray
constexpr int CH   = 4;         // samples per chunk
constexpr int NCH  = NS / CH;   // 16 chunks
constexpr int BLK  = 256;       // 8 wave32 per block
constexpr int WAVE = 32;
constexpr int RGBF = CH * 3;    // 12 floats (48 B) of rgb per ray per chunk

__global__ __launch_bounds__(BLK) void nerf_render(
    const float* __restrict__ rgb, const float* __restrict__ sigma,
    const float* __restrict__ tv, float* __restrict__ image,
    float* __restrict__ depth, float* __restrict__ weights, int nrays)
{
#if USE_TDM
  // per-wave 32 rays x 48 B rgb tile, double buffered: 2 x 12 KB per block
  __shared__ __align__(16) float rgbuf[2][BLK * RGBF];
#endif
  const int tid  = threadIdx.x;
  const int ray  = blockIdx.x * BLK + tid;
  const int cray = ray < nrays ? ray : nrays - 1;

  const float* tp = tv      + (size_t)cray * NS;
  const float* sp = sigma   + (size_t)cray * NS;
  float*       wp = weights + (size_t)cray * NS;

#if USE_TDM
  const int widx = tid >> 5;                       // wave index in block
  int wbase = blockIdx.x * BLK + (tid & ~(WAVE - 1));
  if (wbase > nrays - WAVE) wbase = nrays - WAVE;  // keep whole-wave tile in bounds
  const unsigned long long rgbAddr =
      (unsigned long long)(uintptr_t)rgb + (unsigned long long)wbase * (NS * 3 * 4);

  auto tdm_issue = [&](int c, int buf) {
    // Tile: 32 rows (rays) x 12 fp32; row stride = 192 fp32 (one ray).
    unsigned long long ga = rgbAddr + (unsigned)(c * RGBF * 4);
    unsigned ldsoff = (unsigned)(uintptr_t)&rgbuf[buf][widx * WAVE * RGBF];
    u32x4 g0 = { 1u,                                        // count=1, user D#
                 ldsoff,                                    // lds_addr [63:32]
                 (unsigned)ga,                              // global_addr lo
                 (unsigned)((ga >> 32) & 0x01FFFFFFu)       // global_addr [56:32]
                   | (2u << 30) };                          // type=2 ("image")
    i32x8 g1 = { 0x00020000,                                // wg_mask=0, data_size=4B
                 RGBF << 16,                                // tensor_dim0 = 12
                 WAVE << 16,                                // tensor_dim1 = 32
                 RGBF << 16,                                // tile_dim0   = 12
                 WAVE,                                      // tile_dim1   = 32
                 NS * 3,                                    // dim0_stride = 192 elems
                 0, 0 };
    i32x4 gz4 = { 0, 0, 0, 0 };                             // 2-D: groups 2/3 unused
    i32x8 gz8 = { 0, 0, 0, 0, 0, 0, 0, 0 };                 // extra group (clang-23 form)
    __builtin_amdgcn_tensor_load_to_lds(g0, g1, gz4, gz4, gz8, 0);
  };
  tdm_issue(0, 0);
#else
  const float* rp = rgb + (size_t)cray * (NS * 3);
#endif

  float T = 1.0f, ir = 0.f, ig = 0.f, ib = 0.f, dep = 0.f;

  for (int c = 0; c < NCH; ++c) {
    const int buf = c & 1;
#if USE_TDM
    if (c + 1 < NCH) {           // prefetch next chunk, then wait for current
      tdm_issue(c + 1, buf ^ 1);
      __builtin_amdgcn_s_wait_tensorcnt(1);
    } else {
      __builtin_amdgcn_s_wait_tensorcnt(0);
    }
    const f32x4* rc = (const f32x4*)&rgbuf[buf][(size_t)tid * RGBF];
    f32x4 r0 = rc[0], r1 = rc[1], r2 = rc[2];
#else
    const f32x4* rc = (const f32x4*)(rp + c * RGBF);
    f32x4 r0 = __builtin_nontemporal_load(rc + 0);
    f32x4 r1 = __builtin_nontemporal_load(rc + 1);
    f32x4 r2 = __builtin_nontemporal_load(rc + 2);
#endif
    f32x4 tc = *(const f32x4*)(tp + c * CH);
    f32x4 sg = __builtin_nontemporal_load((const f32x4*)(sp + c * CH));
    float tn = (c + 1 < NCH) ? tp[c * CH + 4] : 0.f;    // one-sample lookahead

    float d0 = tc.y - tc.x, d1 = tc.z - tc.y, d2 = tc.w - tc.z;
    float d3 = (c + 1 < NCH) ? (tn - tc.w) : FARD;

    float e0 = __expf(-sg.x * d0);
    float e1 = __expf(-sg.y * d1);
    float e2 = __expf(-sg.z * d2);
    float e3 = __expf(-sg.w * d3);

    float w0 = (1.f - e0) * T; T *= e0 + EPSV;
    float w1 = (1.f - e1) * T; T *= e1 + EPSV;
    float w2 = (1.f - e2) * T; T *= e2 + EPSV;
    float w3 = (1.f - e3) * T; T *= e3 + EPSV;

    ir  += w0 * r0.x + w1 * r0.w + w2 * r1.z + w3 * r2.y;
    ig  += w0 * r0.y + w1 * r1.x + w2 * r1.w + w3 * r2.z;
    ib  += w0 * r0.z + w1 * r1.y + w2 * r2.x + w3 * r2.w;
    dep += w0 * tc.x + w1 * tc.y + w2 * tc.z + w3 * tc.w;

    if (ray < nrays) {
      f32x4 wv = { w0, w1, w2, w3 };
      __builtin_nontemporal_store(wv, (f32x4*)(wp + c * CH));
    }
  }

  if (ray < nrays) {
    image[(size_t)ray * 3 + 0] = ir;
    image[(size_t)ray * 3 + 1] = ig;
    image[(size_t)ray * 3 + 2] = ib;
    depth[ray] = dep;
  }
}

extern "C" void kernel_launch(void* const* d_in, const int* in_sizes, int n_in,
                              void* d_out, int out_size, void* d_ws, size_t ws_size,
                              hipStream_t stream) {
  (void)n_in; (void)out_size; (void)d_ws; (void)ws_size;
  const float* rgb   = (const float*)d_in[0];
  const float* sigma = (const float*)d_in[1];
  const float* tvals = (const float*)d_in[2];
  const int nrays = in_sizes[2] / NS;          // B*H*W

  float* out     = (float*)d_out;              // [image | depth | weights]
  float* image   = out;
  float* depth   = out + (size_t)nrays * 3;
  float* weights = depth + nrays;

  const int grid = (nrays + BLK - 1) / BLK;
  nerf_render<<<grid, BLK, 0, stream>>>(rgb, sigma, tvals, image, depth, weights, nrays);
}